// BilinearSampler_18837726560482
// MI455X (gfx1250) — compile-verified
//
#include <hip/hip_runtime.h>

// Bilinear grid sample, NHWC f32.  B=16, H=256, W=256, C=32.
//
// Mapping: 8 lanes per output pixel, each lane owns 4 channels (float4).
//  - each neighbor gather is a coalesced 128B burst (one pixel = 32 ch = 128B)
//  - output store is a perfectly linear float4 stream: out4[tid]
// Memory policy: NT stores for output / NT loads for coords so the 128MB image
// stays resident in the 192MB L2 (each input pixel is gathered ~4x on average;
// L2 residency turns that amplification into L2 hits, keeping HBM traffic near
// the 264MB floor -> ~11us at 23.3 TB/s).
//
// All offsets are unsigned 32-bit so the backend can use GVS addressing
// (SGPR base + 32-bit VGPR offset) instead of per-lane 64-bit adds.

namespace {
constexpr int Bn = 16;
constexpr int Hn = 256;
constexpr int Wn = 256;
constexpr int Cn = 32;
constexpr unsigned NPIX = Bn * Hn * Wn;                   // 1,048,576 pixels
constexpr unsigned TOTAL_THREADS = NPIX * (Cn / 4);       // 8,388,608
constexpr int BLOCK = 256;                                // 8 wave32s / block
}

typedef __attribute__((ext_vector_type(4))) float v4f;

__global__ __launch_bounds__(BLOCK) void bilinear_sample_kernel(
    const float* __restrict__ img,
    const float* __restrict__ xs,
    const float* __restrict__ ys,
    float* __restrict__ out)
{
    const unsigned tid = blockIdx.x * (unsigned)BLOCK + threadIdx.x;
    const unsigned p   = tid >> 3;          // pixel index in [0, B*H*W)
    const unsigned c4  = (tid & 7u) << 2;   // channel offset: 0,4,...,28

    // Stream the coordinate arrays 2048 pixels (~64 blocks) ahead. Unconditional: the 8
    // lanes of a pixel-group hit the same cacheline and merge in the
    // coalescer; a guard costs more (exec save/restore + dup address math).
    // Speculative prefetch past the end is silently dropped by HW.
    __builtin_prefetch(xs + p + 2048, 0, 0);   // -> global_prefetch_b8
    __builtin_prefetch(ys + p + 2048, 0, 0);

    // Coordinates are read exactly once across the grid -> non-temporal loads
    // (don't spend L2 capacity on them).
    const float xsv = __builtin_nontemporal_load(xs + p);
    const float ysv = __builtin_nontemporal_load(ys + p);

    // Map [-1,1] -> pixel coords, same expression order as the reference.
    const float x = 0.5f * (xsv + 1.0f) * (float)(Wn - 1);
    const float y = 0.5f * (ysv + 1.0f) * (float)(Hn - 1);

    // x,y >= 0 (x_s,y_s are uniform in [-1,1]), so trunc == floor:
    // a single v_cvt_i32_f32 instead of v_floor_f32 + cvt.
    const int x0 = (int)x;
    const int y0 = (int)y;
    const int x1 = x0 + 1;
    const int y1 = y0 + 1;

    const int x0c = min(max(x0, 0), Wn - 1);
    const int x1c = min(max(x1, 0), Wn - 1);
    const int y0c = min(max(y0, 0), Hn - 1);
    const int y1c = min(max(y1, 0), Hn - 1);

    // Reference computes the weights from the *clamped* integer coords.
    const float x0f = (float)x0c, x1f = (float)x1c;
    const float y0f = (float)y0c, y1f = (float)y1c;
    const float wa = (x1f - x) * (y1f - y);   // top-left
    const float wb = (x1f - x) * (y - y0f);   // bottom-left
    const float wc = (x - x0f) * (y1f - y);   // top-right
    const float wd = (x - x0f) * (y - y0f);   // bottom-right

    // b*H*W term: H*W == 65536, so it's just the top bits of p.
    const unsigned bbase = (p >> 16) << 16;

    // Unsigned element offsets (max = NPIX*C = 2^25, fits easily).
    const unsigned r0 = bbase + (unsigned)(y0c * Wn);
    const unsigned r1 = bbase + (unsigned)(y1c * Wn);
    const unsigned i00 = ((r0 + (unsigned)x0c) << 5) + c4;  // *C = <<5
    const unsigned i10 = ((r1 + (unsigned)x0c) << 5) + c4;
    const unsigned i01 = ((r0 + (unsigned)x1c) << 5) + c4;
    const unsigned i11 = ((r1 + (unsigned)x1c) << 5) + c4;

    // 4 coalesced 128B gathers per pixel-group (global_load_b128 per lane),
    // default temporal policy -> these stay L2-resident.
    const v4f Ia = *reinterpret_cast<const v4f*>(img + i00);
    const v4f Ib = *reinterpret_cast<const v4f*>(img + i10);
    const v4f Ic = *reinterpret_cast<const v4f*>(img + i01);
    const v4f Id = *reinterpret_cast<const v4f*>(img + i11);

    v4f r;
    r.x = fmaf(wd, Id.x, fmaf(wc, Ic.x, fmaf(wb, Ib.x, wa * Ia.x)));
    r.y = fmaf(wd, Id.y, fmaf(wc, Ic.y, fmaf(wb, Ib.y, wa * Ia.y)));
    r.z = fmaf(wd, Id.z, fmaf(wc, Ic.z, fmaf(wb, Ib.z, wa * Ia.z)));
    r.w = fmaf(wd, Id.w, fmaf(wc, Ic.w, fmaf(wb, Ib.w, wa * Ia.w)));

    // out offset = p*C + c4 = tid*4 -> linear float4 stream.
    // Non-temporal: the 128MB output must not evict the L2-resident image.
    __builtin_nontemporal_store(r, reinterpret_cast<v4f*>(out) + tid);
}

extern "C" void kernel_launch(void* const* d_in, const int* in_sizes, int n_in,
                              void* d_out, int out_size, void* d_ws, size_t ws_size,
                              hipStream_t stream) {
    const float* img = (const float*)d_in[0];   // batch_inputs [16,256,256,32]
    const float* xs  = (const float*)d_in[1];   // x_s [16,256,256]
    const float* ys  = (const float*)d_in[2];   // y_s [16,256,256]
    float* out = (float*)d_out;                  // [16,256,256,32]

    const unsigned grid = TOTAL_THREADS / BLOCK; // 32768 blocks
    bilinear_sample_kernel<<<grid, BLOCK, 0, stream>>>(img, xs, ys, out);
}